// TripletLossWithHardMining_52690658787917
// MI455X (gfx1250) — compile-verified
//
#include <hip/hip_runtime.h>
#include <hip/hip_bf16.h>
#include <math.h>

// Problem constants (match reference)
#define BDIM 256
#define NDIM 256
#define DDIM 512
#define MARGIN 0.5f
#define EPS_F 1e-6f

#define WAVES_PER_BLOCK 8
#define ROWS (BDIM * NDIM)                 // 65536
#define SLICE ((size_t)NDIM * DDIM)        // 131072 floats per [N,D] slice
#define BND ((size_t)BDIM * SLICE)         // 33554432 floats per output tensor

// ---------------------------------------------------------------------------
// Phase 1: per-row L2 distances, streamed through LDS with the gfx1250 async
// global->LDS path (double-buffered, 4 chunks of 128 floats per tensor).
// One wave (32 lanes) owns one row; each lane handles 4 floats per tensor per
// chunk (float4 in LDS).
// ---------------------------------------------------------------------------
__device__ __forceinline__ void async_issue_chunk(const float* a, const float* p,
                                                  const float* n, int c, int lane,
                                                  uint32_t lds_base) {
  // chunk c covers floats [c*128, c*128+128); lane loads its float4 at +lane*4
  uint64_t ga = (uint64_t)(uintptr_t)(a + c * 128 + lane * 4);
  uint64_t gp = (uint64_t)(uintptr_t)(p + c * 128 + lane * 4);
  uint64_t gn = (uint64_t)(uintptr_t)(n + c * 128 + lane * 4);
  uint32_t la = lds_base;                 // tensor 0 (anchor)
  uint32_t lp = lds_base + 32 * 16;       // tensor 1 (positive)
  uint32_t ln = lds_base + 2 * 32 * 16;   // tensor 2 (negative)
  asm volatile("global_load_async_to_lds_b128 %0, %1, off" :: "v"(la), "v"(ga) : "memory");
  asm volatile("global_load_async_to_lds_b128 %0, %1, off" :: "v"(lp), "v"(gp) : "memory");
  asm volatile("global_load_async_to_lds_b128 %0, %1, off" :: "v"(ln), "v"(gn) : "memory");
}

__device__ __forceinline__ void accum_chunk(const float4* a4, const float4* p4,
                                            const float4* n4, float& sp, float& sn) {
  float4 a = *a4;   // ds_load_b128
  float4 p = *p4;
  float4 q = *n4;
  float d;
  d = a.x - p.x + EPS_F; sp += d * d;
  d = a.y - p.y + EPS_F; sp += d * d;
  d = a.z - p.z + EPS_F; sp += d * d;
  d = a.w - p.w + EPS_F; sp += d * d;
  d = a.x - q.x + EPS_F; sn += d * d;
  d = a.y - q.y + EPS_F; sn += d * d;
  d = a.z - q.z + EPS_F; sn += d * d;
  d = a.w - q.w + EPS_F; sn += d * d;
}

__global__ void __launch_bounds__(256)
dist_kernel(const float* __restrict__ A, const float* __restrict__ P,
            const float* __restrict__ Ng, float* __restrict__ dpos,
            float* __restrict__ dneg) {
  // [wave][parity][tensor][lane] : 8*2*3*32*16B = 24 KB LDS
  __shared__ float4 smem[WAVES_PER_BLOCK][2][3][32];

  const int tid  = threadIdx.x;
  const int w    = tid >> 5;
  const int lane = tid & 31;
  const int row  = blockIdx.x * WAVES_PER_BLOCK + w;   // exactly ROWS rows launched
  const size_t base = (size_t)row * DDIM;

  const float* a = A + base;
  const float* p = P + base;
  const float* n = Ng + base;

  const uint32_t lds0 = (uint32_t)(uintptr_t)&smem[w][0][0][lane];
  const uint32_t lds1 = (uint32_t)(uintptr_t)&smem[w][1][0][lane];

  float sp = 0.f, sn = 0.f;

  // 2-deep software pipeline over 4 chunks. 3 async loads per stage;
  // s_wait_asynccnt 3 guarantees the older stage's chunk has landed in LDS.
  async_issue_chunk(a, p, n, 0, lane, lds0);
  async_issue_chunk(a, p, n, 1, lane, lds1);
  asm volatile("s_wait_asynccnt 3" ::: "memory");
  accum_chunk(&smem[w][0][0][lane], &smem[w][0][1][lane], &smem[w][0][2][lane], sp, sn);

  async_issue_chunk(a, p, n, 2, lane, lds0);
  asm volatile("s_wait_asynccnt 3" ::: "memory");
  accum_chunk(&smem[w][1][0][lane], &smem[w][1][1][lane], &smem[w][1][2][lane], sp, sn);

  async_issue_chunk(a, p, n, 3, lane, lds1);
  asm volatile("s_wait_asynccnt 3" ::: "memory");
  accum_chunk(&smem[w][0][0][lane], &smem[w][0][1][lane], &smem[w][0][2][lane], sp, sn);

  asm volatile("s_wait_asynccnt 0" ::: "memory");
  accum_chunk(&smem[w][1][0][lane], &smem[w][1][1][lane], &smem[w][1][2][lane], sp, sn);

  // wave32 butterfly reduction
  #pragma unroll
  for (int off = 16; off > 0; off >>= 1) {
    sp += __shfl_xor(sp, off, 32);
    sn += __shfl_xor(sn, off, 32);
  }
  if (lane == 0) {
    dpos[row] = sqrtf(sp);
    dneg[row] = sqrtf(sn);
  }
}

// ---------------------------------------------------------------------------
// Phase 2: per-batch hard mining. One 256-thread block per batch.
// First-occurrence tie-break matches jnp.argmax/argmin (all -inf -> idx 0).
// ---------------------------------------------------------------------------
__global__ void __launch_bounds__(256)
mine_kernel(const float* __restrict__ dpos, const float* __restrict__ dneg,
            int* __restrict__ neg_idx, int* __restrict__ pos_idx,
            float* __restrict__ partial, float* __restrict__ out_valid_neg,
            float* __restrict__ out_valid_pos) {
  const int b = blockIdx.x;
  const int n = threadIdx.x;

  const float dp = dpos[b * NDIM + n];
  const float dn = dneg[b * NDIM + n];
  const bool mneg = (dp - dn + MARGIN) > 0.f;
  const bool mpos = (dn - dp + MARGIN) > 0.f;

  __shared__ float s1[NDIM]; __shared__ int i1[NDIM];   // argmax of dn among mneg
  __shared__ float s2[NDIM]; __shared__ int i2[NDIM];   // argmin of dp among mpos
  __shared__ float sl[NDIM];                            // loss partial

  s1[n] = mneg ? dn : -INFINITY; i1[n] = n;
  s2[n] = mpos ? dp :  INFINITY; i2[n] = n;
  const float t = MARGIN + dp - dn;
  sl[n] = t > 0.f ? t : 0.f;
  __syncthreads();

  for (int off = 128; off > 0; off >>= 1) {
    if (n < off) {
      const float c1 = s1[n], o1 = s1[n + off];
      if (o1 > c1 || (o1 == c1 && i1[n + off] < i1[n])) { s1[n] = o1; i1[n] = i1[n + off]; }
      const float c2 = s2[n], o2 = s2[n + off];
      if (o2 < c2 || (o2 == c2 && i2[n + off] < i2[n])) { s2[n] = o2; i2[n] = i2[n + off]; }
      sl[n] += sl[n + off];
    }
    __syncthreads();
  }

  if (n == 0) {
    const bool vneg = (s1[0] != -INFINITY);
    const bool vpos = (s2[0] !=  INFINITY);
    neg_idx[b] = vneg ? i1[0] : -1;
    pos_idx[b] = vpos ? i2[0] : -1;
    out_valid_neg[b] = vneg ? 1.0f : 0.0f;
    out_valid_pos[b] = vpos ? 1.0f : 0.0f;
    partial[b] = sl[0];
  }
}

// ---------------------------------------------------------------------------
// Phase 3: finalize the scalar loss (deterministic, no atomics).
// ---------------------------------------------------------------------------
__global__ void __launch_bounds__(256)
finalize_kernel(const float* __restrict__ partial, float* __restrict__ out_loss) {
  __shared__ float s[BDIM];
  const int n = threadIdx.x;
  s[n] = partial[n];
  __syncthreads();
  for (int off = 128; off > 0; off >>= 1) {
    if (n < off) s[n] += s[n + off];
    __syncthreads();
  }
  if (n == 0) out_loss[0] = s[0] * (1.0f / (float)(BDIM * NDIM));
}

// ---------------------------------------------------------------------------
// Phase 4: gather/zero the hard_neg / hard_pos slices.
// Source loads are 16B-aligned float4; destinations in d_out start at float
// offset 1 (4B-aligned only), so use scalar non-temporal b32 stores — the
// 256 MB write-once stream should not evict L2 lines holding the (repeating)
// gathered source slices.
// grid = (64 chunks, B, 2 tensors), block = 256; each thread moves 8 floats.
// ---------------------------------------------------------------------------
__global__ void __launch_bounds__(256)
gather_kernel(const float* __restrict__ Pos, const float* __restrict__ Neg,
              const int* __restrict__ neg_idx, const int* __restrict__ pos_idx,
              float* __restrict__ out) {
  const int z     = blockIdx.z;        // 0 = hard_neg, 1 = hard_pos
  const int b     = blockIdx.y;
  const int chunk = blockIdx.x;        // 0..63
  const int t     = threadIdx.x;

  const int idx = (z == 0) ? neg_idx[b] : pos_idx[b];
  const float* src = (z == 0) ? Neg : Pos;

  const size_t off0 = (size_t)chunk * 2048 + (size_t)t * 4;
  const size_t off1 = off0 + 1024;
  float* d0 = out + 1 + (size_t)z * BND + (size_t)b * SLICE + off0;
  float* d1 = d0 + 1024;

  if (idx < 0) {
    __builtin_nontemporal_store(0.0f, d0 + 0);
    __builtin_nontemporal_store(0.0f, d0 + 1);
    __builtin_nontemporal_store(0.0f, d0 + 2);
    __builtin_nontemporal_store(0.0f, d0 + 3);
    __builtin_nontemporal_store(0.0f, d1 + 0);
    __builtin_nontemporal_store(0.0f, d1 + 1);
    __builtin_nontemporal_store(0.0f, d1 + 2);
    __builtin_nontemporal_store(0.0f, d1 + 3);
  } else {
    const float* s = src + (size_t)idx * SLICE;
    const float4 v0 = *(const float4*)(s + off0);
    const float4 v1 = *(const float4*)(s + off1);
    __builtin_nontemporal_store(v0.x, d0 + 0);
    __builtin_nontemporal_store(v0.y, d0 + 1);
    __builtin_nontemporal_store(v0.z, d0 + 2);
    __builtin_nontemporal_store(v0.w, d0 + 3);
    __builtin_nontemporal_store(v1.x, d1 + 0);
    __builtin_nontemporal_store(v1.y, d1 + 1);
    __builtin_nontemporal_store(v1.z, d1 + 2);
    __builtin_nontemporal_store(v1.w, d1 + 3);
  }
}

// ---------------------------------------------------------------------------
// d_out layout (floats, reference return order):
//   [0]                      loss
//   [1 .. 1+BND)             hard_neg
//   [1+BND .. 1+2*BND)       hard_pos
//   [1+2*BND .. +B)          valid_neg (0/1)
//   [1+2*BND+B .. +B)        valid_pos (0/1)
// d_ws layout (floats/ints):
//   dist_pos[ROWS], dist_neg[ROWS], neg_idx[B] (int), pos_idx[B] (int),
//   partial[B]   -> ~516 KB
// ---------------------------------------------------------------------------
extern "C" void kernel_launch(void* const* d_in, const int* in_sizes, int n_in,
                              void* d_out, int out_size, void* d_ws, size_t ws_size,
                              hipStream_t stream) {
  const float* A  = (const float*)d_in[0];
  const float* P  = (const float*)d_in[1];
  const float* Ng = (const float*)d_in[2];
  float* out = (float*)d_out;

  float* dpos    = (float*)d_ws;
  float* dneg    = dpos + ROWS;
  int*   nidx    = (int*)(dneg + ROWS);
  int*   pidx    = nidx + BDIM;
  float* partial = (float*)(pidx + BDIM);

  float* out_loss      = out;
  float* out_valid_neg = out + 1 + 2 * BND;
  float* out_valid_pos = out_valid_neg + BDIM;

  dist_kernel<<<ROWS / WAVES_PER_BLOCK, 256, 0, stream>>>(A, P, Ng, dpos, dneg);
  mine_kernel<<<BDIM, 256, 0, stream>>>(dpos, dneg, nidx, pidx, partial,
                                        out_valid_neg, out_valid_pos);
  finalize_kernel<<<1, 256, 0, stream>>>(partial, out_loss);
  gather_kernel<<<dim3(64, BDIM, 2), 256, 0, stream>>>(P, Ng, nidx, pidx, out);
}